// GATV2_64836826300770
// MI455X (gfx1250) — compile-verified
//
#include <hip/hip_runtime.h>
#include <hip/hip_bf16.h>
#include <math.h>

// ---------------------------------------------------------------------------
// GATv2 (2 layers) for MI455X / gfx1250.
//  - GEMMs via V_WMMA_F32_16X16X4_F32 (exact fp32 matrix path on CDNA5)
//  - A-tiles staged into LDS by the Tensor Data Mover (TENSOR_LOAD_TO_LDS),
//    with hardware row padding for bank-conflict-free WMMA fragment reads
//  - Edge phase: segment-softmax via f32 global atomics (L2-resident tables)
// Workspace requirement: ~183 MB (see offsets below).
// ---------------------------------------------------------------------------

typedef __attribute__((ext_vector_type(2))) float v2f;
typedef __attribute__((ext_vector_type(8))) float v8f;
typedef __attribute__((ext_vector_type(4))) unsigned int u32x4;
typedef __attribute__((ext_vector_type(8))) int i32x8;
typedef __attribute__((ext_vector_type(4))) int i32x4;

#define NN     50000
#define EE     800000
#define E2T    (EE + NN)      // edges + self loops = 850000
#define FIN    128
#define HID    64
#define HEADS  4
#define C1     256            // HEADS*HID
#define C1X    512            // [xl | xr] concat width (layer 1)
#define NCLS   16
#define C2X    32             // [xl2 | xr2] concat width (layer 2)
#define SLOPE  0.2f

// -------- workspace layout (float offsets) --------
static const size_t OFF_XLR1  = 0;                              // NN*512
static const size_t OFF_E1    = OFF_XLR1  + (size_t)NN * C1X;   // E2T*4
static const size_t OFF_EMAX1 = OFF_E1    + (size_t)E2T * HEADS;// NN*4
static const size_t OFF_DEN1  = OFF_EMAX1 + (size_t)NN * HEADS; // NN*4
static const size_t OFF_H1    = OFF_DEN1  + (size_t)NN * HEADS; // NN*256 (acc -> h, in place)
static const size_t OFF_XLR2  = OFF_H1    + (size_t)NN * C1;    // NN*32
static const size_t OFF_E2    = OFF_XLR2  + (size_t)NN * C2X;   // E2T
static const size_t OFF_EMAX2 = OFF_E2    + (size_t)E2T;        // NN
static const size_t OFF_DEN2  = OFF_EMAX2 + (size_t)NN;         // NN
static const size_t OFF_O2    = OFF_DEN2  + (size_t)NN;         // NN*16

// Publish a TDM-written LDS region to the compiler: the address operand makes
// the LDS array *escape*, so GlobalsAA can no longer prove "never stored" and
// fold its loads to undef; the memory clobber orders subsequent reads.
__device__ __forceinline__ void tdm_lds_publish(const void* lds_ptr) {
    asm volatile("" :: "v"(lds_ptr) : "memory");
}

// ---------------------------------------------------------------------------
// Tensor Data Mover: 2-D tile load Global -> LDS with hardware row padding.
// D# bit packing per CDNA5 ISA (group0) / §8.4 (group1).
//   pad_int_code: dwords-per-interval = 2<<code  (6 -> 128 dw, 7 -> 256 dw)
//   pad_amt_code: pad dwords = code+1            (3 -> 4 dw)
// This toolchain uses the 6-arg builtin variant:
//   (u32x4 g0, i32x8 g1, i32x4 g2, i32x4 g3, i32x8 g4, i32 cpol)
__device__ __forceinline__ void tdm_load_tile_2d(
    unsigned lds_byte_off, const float* gptr,
    unsigned tensor_d0, unsigned tensor_d1,
    unsigned tile_d0,   unsigned tile_d1,
    unsigned stride_d0, unsigned pad_int_code, unsigned pad_amt_code)
{
    const unsigned long long ga = (unsigned long long)(uintptr_t)gptr;
    u32x4 g0;
    g0[0] = 1u;                                   // count=1 (valid user D#)
    g0[1] = lds_byte_off;                         // lds_addr
    g0[2] = (unsigned)(ga & 0xFFFFFFFFu);         // global_addr[31:0]
    g0[3] = ((unsigned)(ga >> 32) & 0x01FFFFFFu)  // global_addr[56:32]
            | (2u << 30);                         // type = 2 ("image")
    i32x8 g1;
    g1[0] = (int)((2u << 16)                      // data_size = 4 bytes
                | (1u << 20)                      // pad_enable
                | (pad_int_code << 22)
                | (pad_amt_code << 25));
    g1[1] = (int)((tensor_d0 & 0xFFFFu) << 16);                 // dim0[15:0] @ bits 63:48
    g1[2] = (int)(((tensor_d0 >> 16) & 0xFFFFu)                 // dim0[31:16]
                | ((tensor_d1 & 0xFFFFu) << 16));               // dim1[15:0]
    g1[3] = (int)(((tensor_d1 >> 16) & 0xFFFFu)                 // dim1[31:16]
                | ((tile_d0 & 0xFFFFu) << 16));                 // tile_dim0
    g1[4] = (int)(tile_d1 & 0xFFFFu);                           // tile_dim1 (tile_dim2=0)
    g1[5] = (int)stride_d0;                                     // dim0_stride[31:0]
    g1[6] = 0;                                                  // dim0_stride[47:32], dim1_stride lo
    g1[7] = 0;
    const i32x4 z4 = {0, 0, 0, 0};
    const i32x8 z8 = {0, 0, 0, 0, 0, 0, 0, 0};
    __builtin_amdgcn_tensor_load_to_lds(g0, g1, z4, z4, z8, 0);
}

// ---------------------------------------------------------------------------
__global__ void k_fill(float* __restrict__ p, float v, int n) {
    int i = blockIdx.x * blockDim.x + threadIdx.x;
    if (i < n) p[i] = v;
}

// ---------------------------------------------------------------------------
// GEMM1: out[N,512] = x[N,128] @ [Wl1 | Wr1]   (Wl1,Wr1: 128x256 row-major)
// Block = 1024 threads (32 waves). TDM stages one 16x128 strip of x into LDS
// (padded to 132-dword rows -> conflict-free column reads). Each wave owns a
// 16-wide column tile; K-loop of 32 f32 WMMAs.
__global__ __launch_bounds__(1024)
void k_gemm1(const float* __restrict__ x, const float* __restrict__ Wl,
             const float* __restrict__ Wr, float* __restrict__ out) {
    __shared__ float sx[16][FIN + 4];       // TDM pads each 128-dw row by 4 dw
    const int row0 = blockIdx.x * 16;
    if (threadIdx.x < 32) {                 // wave 0 issues the tile DMA
        tdm_load_tile_2d(0u, x + (size_t)row0 * FIN,
                         FIN, NN, FIN, 16, FIN,
                         /*interval 128dw*/6, /*pad 4dw*/3);
        __builtin_amdgcn_s_wait_tensorcnt(0);
    }
    __syncthreads();
    tdm_lds_publish(&sx[0][0]);             // escape LDS array + memory clobber

    const int wave = threadIdx.x >> 5;      // 0..31
    const int lane = threadIdx.x & 31;
    const int colb = wave * 16;             // 0..496
    const float* Wp = (colb < 256) ? Wl : Wr;
    const int cb   = colb & 255;
    const int l16  = lane & 15;
    const int koff = (lane >> 4) << 1;      // lanes 16-31 carry K+2,K+3
    const int hi   = lane >> 4;

    v8f acc = {};
    for (int k = 0; k < FIN; k += 4) {
        v2f a, b;
        a.x = sx[l16][k + koff];
        a.y = sx[l16][k + koff + 1];
        b.x = Wp[(size_t)(k + koff)     * 256 + cb + l16];
        b.y = Wp[(size_t)(k + koff + 1) * 256 + cb + l16];
        acc = __builtin_amdgcn_wmma_f32_16x16x4_f32(
            false, a, false, b, (short)0, acc, false, false);
    }
    const int col = colb + l16;
#pragma unroll
    for (int v = 0; v < 8; ++v)
        out[(size_t)(row0 + v + hi * 8) * C1X + col] = acc[v];
}

// ---------------------------------------------------------------------------
// GEMM2: out[N,32] = h[N,256] @ [Wl2 | Wr2]   (Wl2,Wr2: 256x16 row-major)
__global__ __launch_bounds__(64)
void k_gemm2(const float* __restrict__ h, const float* __restrict__ Wl,
             const float* __restrict__ Wr, float* __restrict__ out) {
    __shared__ float sh[16][C1 + 4];        // TDM pads each 256-dw row by 4 dw
    const int row0 = blockIdx.x * 16;
    if (threadIdx.x < 32) {
        tdm_load_tile_2d(0u, h + (size_t)row0 * C1,
                         C1, NN, C1, 16, C1,
                         /*interval 256dw*/7, /*pad 4dw*/3);
        __builtin_amdgcn_s_wait_tensorcnt(0);
    }
    __syncthreads();
    tdm_lds_publish(&sh[0][0]);             // escape LDS array + memory clobber

    const int wave = threadIdx.x >> 5;      // 0 -> Wl2, 1 -> Wr2
    const int lane = threadIdx.x & 31;
    const float* Wp = wave ? Wr : Wl;
    const int l16 = lane & 15, koff = (lane >> 4) << 1, hi = lane >> 4;

    v8f acc = {};
    for (int k = 0; k < C1; k += 4) {
        v2f a, b;
        a.x = sh[l16][k + koff];
        a.y = sh[l16][k + koff + 1];
        b.x = Wp[(size_t)(k + koff)     * NCLS + l16];
        b.y = Wp[(size_t)(k + koff + 1) * NCLS + l16];
        acc = __builtin_amdgcn_wmma_f32_16x16x4_f32(
            false, a, false, b, (short)0, acc, false, false);
    }
    const int col = wave * 16 + l16;
#pragma unroll
    for (int v = 0; v < 8; ++v)
        out[(size_t)(row0 + v + hi * 8) * C2X + col] = acc[v];
}

// ---------------------------------------------------------------------------
__device__ __forceinline__ void edge_sd(const int* __restrict__ ei, int eid,
                                        int& s, int& d) {
    if (eid < EE) { s = ei[eid]; d = ei[EE + eid]; }
    else          { s = d = eid - EE; }
}

// Layer-1 edge logits: e[edge,h] = sum_c lrelu(xl[s]+xr[d]) * att ; atomicMax
// One wave per edge; 2 channels/lane per head; shuffle reduce.
__global__ void k_logits1(const int* __restrict__ ei, const float* __restrict__ xlr,
                          const float* __restrict__ att, float* __restrict__ e1,
                          float* __restrict__ emax) {
    const int eid = blockIdx.x * 8 + (threadIdx.x >> 5);
    if (eid >= E2T) return;
    const int lane = threadIdx.x & 31;
    int s, d; edge_sd(ei, eid, s, d);
    const float* xl = xlr + (size_t)s * C1X;
    const float* xr = xlr + (size_t)d * C1X + C1;
#pragma unroll
    for (int h = 0; h < HEADS; ++h) {
        const int c = h * HID + lane * 2;
        const float2 a = *(const float2*)(xl + c);
        const float2 b = *(const float2*)(xr + c);
        const float2 w = *(const float2*)(att + c);
        float m0 = a.x + b.x; m0 = (m0 > 0.f) ? m0 : SLOPE * m0;
        float m1 = a.y + b.y; m1 = (m1 > 0.f) ? m1 : SLOPE * m1;
        float p = m0 * w.x + m1 * w.y;
#pragma unroll
        for (int off = 16; off > 0; off >>= 1) p += __shfl_xor(p, off, 32);
        if (lane == 0) {
            e1[(size_t)eid * HEADS + h] = p;
            atomicMax(&emax[(size_t)d * HEADS + h], p);   // f32 max_num atomic
        }
    }
}

// exp(e - emax[dst]) ; denom accumulation. One thread per (edge, head).
__global__ void k_expden1(const int* __restrict__ ei, const float* __restrict__ emax,
                          float* __restrict__ e1, float* __restrict__ den) {
    int i = blockIdx.x * blockDim.x + threadIdx.x;
    if (i >= E2T * HEADS) return;
    int eid = i >> 2, h = i & 3;
    int s, d; edge_sd(ei, eid, s, d); (void)s;
    float ex = __expf(e1[i] - emax[(size_t)d * HEADS + h]);
    e1[i] = ex;
    atomicAdd(&den[(size_t)d * HEADS + h], ex);
}

// Weighted scatter: acc[dst] += alpha * xl[src]. Wave per edge, 8 ch/lane (float4).
__global__ void k_agg1(const int* __restrict__ ei, const float* __restrict__ xlr,
                       const float* __restrict__ e1, const float* __restrict__ den,
                       float* __restrict__ acc) {
    const int eid = blockIdx.x * 8 + (threadIdx.x >> 5);
    if (eid >= E2T) return;
    const int lane = threadIdx.x & 31;
    int s, d; edge_sd(ei, eid, s, d);
    const int h = lane >> 3;                 // 8 lanes per head block of 64 ch
    const float alpha = e1[(size_t)eid * HEADS + h] /
                        (den[(size_t)d * HEADS + h] + 1e-16f);
    const int c0 = lane * 8;
    const float4 v0 = *(const float4*)(xlr + (size_t)s * C1X + c0);
    const float4 v1 = *(const float4*)(xlr + (size_t)s * C1X + c0 + 4);
    float* ap = acc + (size_t)d * C1 + c0;
    atomicAdd(ap + 0, alpha * v0.x); atomicAdd(ap + 1, alpha * v0.y);
    atomicAdd(ap + 2, alpha * v0.z); atomicAdd(ap + 3, alpha * v0.w);
    atomicAdd(ap + 4, alpha * v1.x); atomicAdd(ap + 5, alpha * v1.y);
    atomicAdd(ap + 6, alpha * v1.z); atomicAdd(ap + 7, alpha * v1.w);
}

// h = elu(acc + b1), in place.
__global__ void k_biaselu(float* __restrict__ h, const float* __restrict__ b) {
    int i = blockIdx.x * blockDim.x + threadIdx.x;
    if (i >= NN * C1) return;
    float v = h[i] + b[i & 255];
    h[i] = (v > 0.f) ? v : (__expf(v) - 1.f);
}

// Layer-2 edge logits (16 classes, 1 head). One thread per edge.
__global__ void k_logits2(const int* __restrict__ ei, const float* __restrict__ xlr2,
                          const float* __restrict__ att2, float* __restrict__ e2,
                          float* __restrict__ emax2) {
    int eid = blockIdx.x * blockDim.x + threadIdx.x;
    if (eid >= E2T) return;
    int s, d; edge_sd(ei, eid, s, d);
    const float* xl = xlr2 + (size_t)s * C2X;
    const float* xr = xlr2 + (size_t)d * C2X + NCLS;
    float p = 0.f;
#pragma unroll
    for (int c = 0; c < NCLS; ++c) {
        float m = xl[c] + xr[c];
        m = (m > 0.f) ? m : SLOPE * m;
        p += m * att2[c];
    }
    e2[eid] = p;
    atomicMax(&emax2[d], p);
}

__global__ void k_expden2(const int* __restrict__ ei, const float* __restrict__ emax2,
                          float* __restrict__ e2, float* __restrict__ den2) {
    int eid = blockIdx.x * blockDim.x + threadIdx.x;
    if (eid >= E2T) return;
    int s, d; edge_sd(ei, eid, s, d); (void)s;
    float ex = __expf(e2[eid] - emax2[d]);
    e2[eid] = ex;
    atomicAdd(&den2[d], ex);
}

// o2[dst,c] += alpha * xl2[src,c]. 16 threads per edge.
__global__ void k_agg2(const int* __restrict__ ei, const float* __restrict__ xlr2,
                       const float* __restrict__ e2, const float* __restrict__ den2,
                       float* __restrict__ o2) {
    int t = blockIdx.x * blockDim.x + threadIdx.x;
    int eid = t >> 4, c = t & 15;
    if (eid >= E2T) return;
    int s, d; edge_sd(ei, eid, s, d);
    float alpha = e2[eid] / (den2[d] + 1e-16f);
    atomicAdd(&o2[(size_t)d * NCLS + c], alpha * xlr2[(size_t)s * C2X + c]);
}

// Final: out = log_softmax(o2 + b2). One thread per node.
__global__ void k_final(const float* __restrict__ o2, const float* __restrict__ b2,
                        float* __restrict__ out) {
    int n = blockIdx.x * blockDim.x + threadIdx.x;
    if (n >= NN) return;
    float v[NCLS];
    float m = -INFINITY;
#pragma unroll
    for (int c = 0; c < NCLS; ++c) {
        v[c] = o2[(size_t)n * NCLS + c] + b2[c];
        m = fmaxf(m, v[c]);
    }
    float sum = 0.f;
#pragma unroll
    for (int c = 0; c < NCLS; ++c) sum += __expf(v[c] - m);
    float lse = m + __logf(sum);
#pragma unroll
    for (int c = 0; c < NCLS; ++c) out[(size_t)n * NCLS + c] = v[c] - lse;
}

// ---------------------------------------------------------------------------
extern "C" void kernel_launch(void* const* d_in, const int* in_sizes, int n_in,
                              void* d_out, int out_size, void* d_ws, size_t ws_size,
                              hipStream_t stream) {
    (void)in_sizes; (void)n_in; (void)out_size; (void)ws_size;
    const float* x    = (const float*)d_in[0];
    const int*   ei   = (const int*)  d_in[1];   // [2,E] src row then dst row
    const float* Wl1  = (const float*)d_in[2];
    const float* Wr1  = (const float*)d_in[3];
    const float* att1 = (const float*)d_in[4];
    const float* b1   = (const float*)d_in[5];
    const float* Wl2  = (const float*)d_in[6];
    const float* Wr2  = (const float*)d_in[7];
    const float* att2 = (const float*)d_in[8];
    const float* b2   = (const float*)d_in[9];

    float* ws    = (float*)d_ws;
    float* xlr1  = ws + OFF_XLR1;
    float* e1    = ws + OFF_E1;
    float* emax1 = ws + OFF_EMAX1;
    float* den1  = ws + OFF_DEN1;
    float* h1    = ws + OFF_H1;     // accumulator, then elu'd features
    float* xlr2  = ws + OFF_XLR2;
    float* e2w   = ws + OFF_E2;
    float* emax2 = ws + OFF_EMAX2;
    float* den2  = ws + OFF_DEN2;
    float* o2    = ws + OFF_O2;

    // deterministic re-init of all accumulators each call
    k_fill<<<(NN * HEADS + 255) / 256, 256, 0, stream>>>(emax1, -INFINITY, NN * HEADS);
    k_fill<<<(NN * HEADS + 255) / 256, 256, 0, stream>>>(den1, 0.f, NN * HEADS);
    k_fill<<<(NN * C1 + 255) / 256, 256, 0, stream>>>(h1, 0.f, NN * C1);
    k_fill<<<(NN + 255) / 256, 256, 0, stream>>>(emax2, -INFINITY, NN);
    k_fill<<<(NN + 255) / 256, 256, 0, stream>>>(den2, 0.f, NN);
    k_fill<<<(NN * NCLS + 255) / 256, 256, 0, stream>>>(o2, 0.f, NN * NCLS);

    // ---- layer 1 ----
    k_gemm1<<<NN / 16, 1024, 0, stream>>>(x, Wl1, Wr1, xlr1);
    k_logits1<<<(E2T + 7) / 8, 256, 0, stream>>>(ei, xlr1, att1, e1, emax1);
    k_expden1<<<(E2T * HEADS + 255) / 256, 256, 0, stream>>>(ei, emax1, e1, den1);
    k_agg1<<<(E2T + 7) / 8, 256, 0, stream>>>(ei, xlr1, e1, den1, h1);
    k_biaselu<<<(NN * C1 + 255) / 256, 256, 0, stream>>>(h1, b1);

    // ---- layer 2 ----
    k_gemm2<<<NN / 16, 64, 0, stream>>>(h1, Wl2, Wr2, xlr2);
    k_logits2<<<(E2T + 255) / 256, 256, 0, stream>>>(ei, xlr2, att2, e2w, emax2);
    k_expden2<<<(E2T + 255) / 256, 256, 0, stream>>>(ei, emax2, e2w, den2);
    k_agg2<<<(E2T * 16 + 255) / 256, 256, 0, stream>>>(ei, xlr2, e2w, den2, o2);
    k_final<<<(NN + 255) / 256, 256, 0, stream>>>(o2, b2, (float*)d_out);
}